// TypedGraphMemory_36051955483062
// MI455X (gfx1250) — compile-verified
//
#include <hip/hip_runtime.h>
#include <math.h>

#define B_ 64
#define D_ 256
#define S_ 1024
#define V_ 1024
#define SPLIT_ 8
#define CH_ (S_ / SPLIT_)   // 128 rows per partial chunk

typedef float v2f __attribute__((ext_vector_type(2)));
typedef float v8f __attribute__((ext_vector_type(8)));
// exact pointee type expected by __builtin_amdgcn_global_load_async_to_lds_b128
typedef int vsi4 __attribute__((vector_size(4 * sizeof(int))));
typedef __attribute__((address_space(1))) vsi4* as1_v4i;
typedef __attribute__((address_space(3))) vsi4* as3_v4i;

__device__ __forceinline__ float gelu_exact(float x) {
    return 0.5f * x * (1.0f + erff(x * 0.70710678118654752440f));
}
__device__ __forceinline__ float sigmoid_(float x) {
    return 1.0f / (1.0f + expf(-x));
}
__device__ __forceinline__ int clampi(int x, int hi) {
    return x < 0 ? 0 : (x > hi ? hi : x);
}

// ---------------------------------------------------------------------------
// Generic f32 WMMA GEMM: C[M,N] = act(A[M,K] @ B[K,N] + bias)
// M = gridDim.y*16, N = gridDim.x*64 (4 waves/block, one 16x16 tile per wave)
// Uses V_WMMA_F32_16X16X4_F32 (exact fp32 semantics).
// ---------------------------------------------------------------------------
__global__ void wmma_gemm_f32_kernel(const float* __restrict__ A, int lda,
                                     const float* __restrict__ Bm, int ldb,
                                     float* __restrict__ C, int ldc,
                                     const float* __restrict__ bias,
                                     int K, int applyGelu) {
    const int lane  = threadIdx.x & 31;
    const int wave  = threadIdx.x >> 5;
    const int m0    = blockIdx.y << 4;
    const int n0    = (blockIdx.x << 6) + (wave << 4);
    const int mrow  = lane & 15;
    const int khalf = (lane >> 4) << 1;          // 0 or 2
    const float* arow = A + (size_t)(m0 + mrow) * lda;
    const int ncol  = n0 + (lane & 15);

    v8f acc = {0.f, 0.f, 0.f, 0.f, 0.f, 0.f, 0.f, 0.f};
    for (int k0 = 0; k0 < K; k0 += 4) {
        v2f a = *(const v2f*)(arow + k0 + khalf);     // one b64 load (K-consecutive)
        v2f b;
        b[0] = Bm[(size_t)(k0 + khalf) * ldb + ncol];
        b[1] = Bm[(size_t)(k0 + khalf + 1) * ldb + ncol];
        acc = __builtin_amdgcn_wmma_f32_16x16x4_f32(
            /*neg_a=*/false, a, /*neg_b=*/false, b,
            /*c_mod=*/(short)0, acc, /*reuse_a=*/false, /*reuse_b=*/false);
    }
    const int mbase = m0 + ((lane >> 4) << 3);   // 0 or 8 within tile
    const float bv = bias ? bias[ncol] : 0.0f;
#pragma unroll
    for (int r = 0; r < 8; ++r) {
        float x = acc[r] + bv;
        if (applyGelu) x = gelu_exact(x);
        C[(size_t)(mbase + r) * ldc + ncol] = x;
    }
}

// ---------------------------------------------------------------------------
// gains[b] = sigmoid(dot(h_mg[b,:], mg_w2) + mg_b2); gains[64+b] = same for sg
// ---------------------------------------------------------------------------
__global__ void gain_kernel(const float* __restrict__ h_mg, const float* __restrict__ h_sg,
                            const float* __restrict__ mg_w2, const float* __restrict__ mg_b2,
                            const float* __restrict__ sg_w2, const float* __restrict__ sg_b2,
                            float* __restrict__ gains) {
    __shared__ float red0[256];
    __shared__ float red1[256];
    const int b = blockIdx.x, t = threadIdx.x;
    red0[t] = h_mg[(size_t)b * D_ + t] * mg_w2[t];
    red1[t] = h_sg[(size_t)b * D_ + t] * sg_w2[t];
    __syncthreads();
    for (int s = 128; s > 0; s >>= 1) {
        if (t < s) { red0[t] += red0[t + s]; red1[t] += red1[t + s]; }
        __syncthreads();
    }
    if (t == 0) {
        gains[b]       = sigmoid_(red0[0] + mg_b2[0]);
        gains[B_ + b]  = sigmoid_(red1[0] + sg_b2[0]);
    }
}

// walk0[b,s] = (s == clamp(query_idx)) * query_valid;  Wsum initialized to walk0
__global__ void walk_init_kernel(float* __restrict__ walk0, float* __restrict__ wsum,
                                 const int* __restrict__ qidx, const float* __restrict__ qvalid) {
    const int b = blockIdx.x, t = threadIdx.x;
    const int qc = clampi(qidx[b], S_ - 1);
    const float qv = qvalid[b];
    for (int i = t; i < S_; i += 256) {
        const float v = (i == qc) ? qv : 0.0f;
        walk0[(size_t)b * S_ + i] = v;
        wsum[(size_t)b * S_ + i]  = v;
    }
}

// ---------------------------------------------------------------------------
// Split-K batched vec(1xS) * mat(Sx1024) partial products.
// grid (B_, SPLIT_); block 256 threads, float4 per thread over 1024 cols.
// Vector chunk (128 floats) staged into LDS via CDNA5 async-to-LDS DMA.
// partial[(b*SPLIT_+ch)*1024 + col] = sum over this chunk's rows.
// ---------------------------------------------------------------------------
__global__ void spmv_partial_kernel(const float* __restrict__ vec,
                                    const float* __restrict__ mat,
                                    float* __restrict__ partial) {
    __shared__ float wsh[CH_];
    const int b = blockIdx.x, ch = blockIdx.y, t = threadIdx.x;
    if (t < CH_ / 4) {
        const float* gsrc = vec + (size_t)b * S_ + ch * CH_ + t * 4;
        __builtin_amdgcn_global_load_async_to_lds_b128(
            (as1_v4i)gsrc, (as3_v4i)&wsh[t * 4], 0, 0);
    }
    asm volatile("s_wait_asynccnt 0x0" ::: "memory");
    __syncthreads();

    const float* base = mat + (size_t)b * S_ * 1024 + (size_t)ch * CH_ * 1024;
    float4 acc = make_float4(0.f, 0.f, 0.f, 0.f);
    for (int s = 0; s < CH_; ++s) {
        const float wv = wsh[s];        // uniform across block
        if (wv == 0.0f) continue;       // sparse-walk fast path
        const float4 m = ((const float4*)(base + (size_t)s * 1024))[t];
        acc.x = fmaf(wv, m.x, acc.x);
        acc.y = fmaf(wv, m.y, acc.y);
        acc.z = fmaf(wv, m.z, acc.z);
        acc.w = fmaf(wv, m.w, acc.w);
    }
    ((float4*)partial)[(size_t)(b * SPLIT_ + ch) * (1024 / 4) + t] = acc;
}

// ---------------------------------------------------------------------------
// Fold SPLIT_ partials -> next walk; add rank-1 step-scatter correction; Wsum +=
// ---------------------------------------------------------------------------
__global__ void walk_reduce_kernel(const float* __restrict__ partial,
                                   const float* __restrict__ prev,
                                   float* __restrict__ next, float* __restrict__ wsum,
                                   const float* __restrict__ gains,
                                   const int* __restrict__ em, const int* __restrict__ src_idx,
                                   const unsigned char* __restrict__ src_valid,
                                   const int* __restrict__ tsym_idx,
                                   const unsigned char* __restrict__ tsym_valid) {
    const int b = blockIdx.x, t = threadIdx.x;
    float4 acc = make_float4(0.f, 0.f, 0.f, 0.f);
#pragma unroll
    for (int ch = 0; ch < SPLIT_; ++ch) {
        const float4 p = ((const float4*)partial)[(size_t)(b * SPLIT_ + ch) * (S_ / 4) + t];
        acc.x += p.x; acc.y += p.y; acc.z += p.z; acc.w += p.w;
    }
    const int e = em[b];
    if (e == 2 && src_valid[b] && tsym_valid[b]) {
        const int sc = clampi(src_idx[b], S_ - 1);
        const int tc = clampi(tsym_idx[b], S_ - 1);
        if ((tc >> 2) == t) {
            const float add = gains[B_ + b] * prev[(size_t)b * S_ + sc];
            if      ((tc & 3) == 0) acc.x += add;
            else if ((tc & 3) == 1) acc.y += add;
            else if ((tc & 3) == 2) acc.z += add;
            else                    acc.w += add;
        }
    }
    const size_t o = (size_t)b * S_ + (size_t)t * 4;
    next[o + 0] = acc.x; next[o + 1] = acc.y; next[o + 2] = acc.z; next[o + 3] = acc.w;
    wsum[o + 0] += acc.x; wsum[o + 1] += acc.y; wsum[o + 2] += acc.z; wsum[o + 3] += acc.w;
}

// ---------------------------------------------------------------------------
// Fold SPLIT_ partials -> acc_values; add rank-1 map-scatter correction
// ---------------------------------------------------------------------------
__global__ void accv_reduce_kernel(const float* __restrict__ partial,
                                   const float* __restrict__ wsum,
                                   float* __restrict__ accv,
                                   const float* __restrict__ gains,
                                   const int* __restrict__ em, const int* __restrict__ src_idx,
                                   const unsigned char* __restrict__ src_valid,
                                   const int* __restrict__ tval_idx,
                                   const unsigned char* __restrict__ tval_valid) {
    const int b = blockIdx.x, t = threadIdx.x;
    float4 acc = make_float4(0.f, 0.f, 0.f, 0.f);
#pragma unroll
    for (int ch = 0; ch < SPLIT_; ++ch) {
        const float4 p = ((const float4*)partial)[(size_t)(b * SPLIT_ + ch) * (V_ / 4) + t];
        acc.x += p.x; acc.y += p.y; acc.z += p.z; acc.w += p.w;
    }
    const int e = em[b];
    if ((e == 0 || e == 1) && src_valid[b] && tval_valid[b]) {
        const int sc = clampi(src_idx[b], S_ - 1);
        const int tc = clampi(tval_idx[b], V_ - 1);
        if ((tc >> 2) == t) {
            const float add = gains[b] * wsum[(size_t)b * S_ + sc];
            if      ((tc & 3) == 0) acc.x += add;
            else if ((tc & 3) == 1) acc.y += add;
            else if ((tc & 3) == 2) acc.z += add;
            else                    acc.w += add;
        }
    }
    const size_t o = (size_t)b * V_ + (size_t)t * 4;
    accv[o + 0] = acc.x; accv[o + 1] = acc.y; accv[o + 2] = acc.z; accv[o + 3] = acc.w;
}

extern "C" void kernel_launch(void* const* d_in, const int* in_sizes, int n_in,
                              void* d_out, int out_size, void* d_ws, size_t ws_size,
                              hipStream_t stream) {
    const float* map_mem      = (const float*)d_in[0];
    const float* step_mem     = (const float*)d_in[1];
    const int*   event_marker = (const int*)d_in[2];
    const int*   source_idx   = (const int*)d_in[3];
    const unsigned char* source_valid = (const unsigned char*)d_in[4];
    const int*   tsym_idx     = (const int*)d_in[5];
    const unsigned char* tsym_valid   = (const unsigned char*)d_in[6];
    const int*   tval_idx     = (const int*)d_in[7];
    const unsigned char* tval_valid   = (const unsigned char*)d_in[8];
    const float* evidence     = (const float*)d_in[9];
    const int*   query_idx    = (const int*)d_in[10];
    const float* query_valid  = (const float*)d_in[11];
    const float* sym_emb      = (const float*)d_in[12];
    const float* val_emb      = (const float*)d_in[13];
    const float* mg_w1 = (const float*)d_in[14];
    const float* mg_b1 = (const float*)d_in[15];
    const float* mg_w2 = (const float*)d_in[16];
    const float* mg_b2 = (const float*)d_in[17];
    const float* sg_w1 = (const float*)d_in[18];
    const float* sg_b1 = (const float*)d_in[19];
    const float* sg_w2 = (const float*)d_in[20];
    const float* sg_b2 = (const float*)d_in[21];
    const float* sf_w1 = (const float*)d_in[22];
    const float* sf_b1 = (const float*)d_in[23];
    const float* sf_w2 = (const float*)d_in[24];
    const float* sf_b2 = (const float*)d_in[25];
    const float* oh_w1 = (const float*)d_in[26];
    const float* oh_b1 = (const float*)d_in[27];
    const float* oh_w2 = (const float*)d_in[28];
    const float* oh_b2 = (const float*)d_in[29];

    // ---- workspace layout (floats) ----
    float* h_mg    = (float*)d_ws;              // 64*256
    float* h_sg    = h_mg    + B_ * D_;         // 64*256
    float* gains   = h_sg    + B_ * D_;         // 2*64
    float* walkA   = gains   + 2 * B_;          // 64*1024
    float* walkB   = walkA   + B_ * S_;         // 64*1024
    float* wsum    = walkB   + B_ * S_;         // 64*1024
    float* accv    = wsum    + B_ * S_;         // 64*1024
    float* gs      = accv    + B_ * V_;         // 64*512
    float* hsf     = gs      + B_ * 2 * D_;     // 64*256
    float* hoh     = hsf     + B_ * D_;         // 64*256
    float* partial = hoh     + B_ * D_;         // 64*8*1024 (2 MB)

    float* out_logits   = (float*)d_out;            // 64*1024
    float* out_feedback = out_logits + B_ * V_;     // 64*256

    const dim3 blk128(128, 1, 1);
    const dim3 spmv_grid(B_, SPLIT_, 1);

    // 1) gate MLP hidden layers: h = gelu(evidence @ w1 + b1)   [64x256 @ 256x256]
    wmma_gemm_f32_kernel<<<dim3(D_ / 64, B_ / 16), blk128, 0, stream>>>(
        evidence, D_, mg_w1, D_, h_mg, D_, mg_b1, D_, 1);
    wmma_gemm_f32_kernel<<<dim3(D_ / 64, B_ / 16), blk128, 0, stream>>>(
        evidence, D_, sg_w1, D_, h_sg, D_, sg_b1, D_, 1);

    // 2) gains[b] = sigmoid(h @ w2 + b2)
    gain_kernel<<<B_, 256, 0, stream>>>(h_mg, h_sg, mg_w2, mg_b2, sg_w2, sg_b2, gains);

    // 3) walk0 = one_hot(query) * query_valid; Wsum = walk0
    walk_init_kernel<<<B_, 256, 0, stream>>>(walkA, wsum, query_idx, query_valid);

    // 4) three graph-walk steps: split-K partials (512 blocks) + fold
    spmv_partial_kernel<<<spmv_grid, 256, 0, stream>>>(walkA, step_mem, partial);
    walk_reduce_kernel<<<B_, 256, 0, stream>>>(partial, walkA, walkB, wsum, gains,
        event_marker, source_idx, source_valid, tsym_idx, tsym_valid);

    spmv_partial_kernel<<<spmv_grid, 256, 0, stream>>>(walkB, step_mem, partial);
    walk_reduce_kernel<<<B_, 256, 0, stream>>>(partial, walkB, walkA, wsum, gains,
        event_marker, source_idx, source_valid, tsym_idx, tsym_valid);

    spmv_partial_kernel<<<spmv_grid, 256, 0, stream>>>(walkA, step_mem, partial);
    walk_reduce_kernel<<<B_, 256, 0, stream>>>(partial, walkA, walkB, wsum, gains,
        event_marker, source_idx, source_valid, tsym_idx, tsym_valid);

    // 5) acc_values = Wsum einsum map_memory (single fused pass, split-K)
    spmv_partial_kernel<<<spmv_grid, 256, 0, stream>>>(wsum, map_mem, partial);
    accv_reduce_kernel<<<B_, 256, 0, stream>>>(partial, wsum, accv, gains,
        event_marker, source_idx, source_valid, tval_idx, tval_valid);

    // 6) graph_state = [Wsum @ sym_emb , acc_values @ val_emb]   (two 64x1024x256 GEMMs)
    wmma_gemm_f32_kernel<<<dim3(D_ / 64, B_ / 16), blk128, 0, stream>>>(
        wsum, S_, sym_emb, D_, gs, 2 * D_, nullptr, S_, 0);
    wmma_gemm_f32_kernel<<<dim3(D_ / 64, B_ / 16), blk128, 0, stream>>>(
        accv, V_, val_emb, D_, gs + D_, 2 * D_, nullptr, V_, 0);

    // 7) feedback = MLP2(gs; sf)   [64x512x256 -> gelu -> 64x256x256]
    wmma_gemm_f32_kernel<<<dim3(D_ / 64, B_ / 16), blk128, 0, stream>>>(
        gs, 2 * D_, sf_w1, D_, hsf, D_, sf_b1, 2 * D_, 1);
    wmma_gemm_f32_kernel<<<dim3(D_ / 64, B_ / 16), blk128, 0, stream>>>(
        hsf, D_, sf_w2, D_, out_feedback, D_, sf_b2, D_, 0);

    // 8) logits = MLP2(gs; oh)     [64x512x256 -> gelu -> 64x256x1024]
    wmma_gemm_f32_kernel<<<dim3(D_ / 64, B_ / 16), blk128, 0, stream>>>(
        gs, 2 * D_, oh_w1, D_, hoh, D_, oh_b1, 2 * D_, 1);
    wmma_gemm_f32_kernel<<<dim3(V_ / 64, B_ / 16), blk128, 0, stream>>>(
        hoh, D_, oh_w2, V_, out_logits, V_, oh_b2, D_, 0);
}